// BiaffineEdgeAttention_66279935312113
// MI455X (gfx1250) — compile-verified
//
#include <hip/hip_runtime.h>
#include <hip/hip_bf16.h>
#include <stdint.h>

typedef float v2f __attribute__((ext_vector_type(2)));
typedef float v8f __attribute__((ext_vector_type(8)));

#define B_DIM 8
#define S_DIM 2048
#define D_DIM 256

#define BK 32
#define LDS_STRIDE 36      // rows of [128 x 32] tiles: 144 B, 16B-aligned, bank-padded
#define KXN_STRIDE 132     // rows of [32 x 128] U tile: 528 B, 16B-aligned, bank-padded
#define TILE_FLOATS 4608   // max(128*36, 32*132)

// ---------------------------------------------------------------------------
// CDNA5 async global->LDS copy (ASYNCcnt-tracked, no VGPR round trip).
// GVS mode: saddr = uniform 64-bit base (SGPR pair), vaddr = per-lane 32-bit
// byte offset, vdst = per-lane LDS byte address.
// ---------------------------------------------------------------------------
__device__ __forceinline__ void async_copy_b128(uint32_t lds_byte_addr,
                                                const float* __restrict__ base,
                                                uint32_t global_byte_off) {
    asm volatile("global_load_async_to_lds_b128 %0, %1, %2"
                 :
                 : "v"(lds_byte_addr), "v"(global_byte_off), "s"(base)
                 : "memory");
}
__device__ __forceinline__ void wait_async_le8() {
    asm volatile("s_wait_asynccnt 0x8" ::: "memory");
}
__device__ __forceinline__ void wait_async_0() {
    asm volatile("s_wait_asynccnt 0x0" ::: "memory");
}
__device__ __forceinline__ uint32_t lds_addr_of(const float* p) {
    // generic pointer to LDS = {aperture_hi32, lds_offset_lo32}
    return (uint32_t)(uintptr_t)p;
}

// ---------------------------------------------------------------------------
// Kernel 1: hs[b,i] = dot(head[b,i,:], wh),  ds[b,o] = dot(dep[b,o,:], wd)
// One wave per output; each lane reduces 8 contiguous floats.
// ---------------------------------------------------------------------------
__global__ __launch_bounds__(256) void rank1_kernel(const float* __restrict__ head,
                                                    const float* __restrict__ dep,
                                                    const float* __restrict__ edge_W,
                                                    float* __restrict__ hs,
                                                    float* __restrict__ ds) {
    const int gwave = (blockIdx.x * blockDim.x + threadIdx.x) >> 5;
    const int lane  = threadIdx.x & 31;
    const int total = B_DIM * S_DIM;

    const float* src;
    const float* w;
    float* dst;
    int idx;
    if (gwave < total) { src = head; w = edge_W;         dst = hs; idx = gwave; }
    else               { src = dep;  w = edge_W + D_DIM; dst = ds; idx = gwave - total; }

    const float* row = src + (size_t)idx * D_DIM + lane * 8;
    const float* wp  = w + lane * 8;

    float4 r0 = *(const float4*)(row);
    float4 r1 = *(const float4*)(row + 4);
    float4 w0 = *(const float4*)(wp);
    float4 w1 = *(const float4*)(wp + 4);

    float sum = r0.x * w0.x + r0.y * w0.y + r0.z * w0.z + r0.w * w0.w
              + r1.x * w1.x + r1.y * w1.y + r1.z * w1.z + r1.w * w1.w;

    #pragma unroll
    for (int off = 16; off > 0; off >>= 1)
        sum += __shfl_xor(sum, off, 32);

    if (lane == 0) dst[idx] = sum;
}

// ---------------------------------------------------------------------------
// fp32 WMMA GEMM: C(M x N) = A(M x K) @ B, 128x128 tile / 256 threads,
// K chunked by 32, double-buffered LDS fed by async global->LDS copies.
//   B_IS_KXN = true : B is K x N row-major (edge_U). LDS keeps Us[k][n];
//                     fragments gather {k,k+1} with stride-KXN_STRIDE reads.
//   B_IS_KXN = false: B is N x K row-major (dep, used as dep^T); LDS Bs[n][k].
//   EPILOGUE : out += hs[row] + ds[col] + bias.
// 8 waves as 4(M) x 2(N); each wave: 2 M-tiles x 4 N-tiles of 16x16 f32.
// ---------------------------------------------------------------------------
template <bool B_IS_KXN, bool EPILOGUE>
__global__ __launch_bounds__(256) void wmma_gemm(const float* __restrict__ Ag,
                                                 const float* __restrict__ Bg,
                                                 float* __restrict__ Cg,
                                                 int M, int N, int K,
                                                 int lda, int ldb, int ldc,
                                                 long strideA, long strideB, long strideC,
                                                 const float* __restrict__ hs,
                                                 const float* __restrict__ ds,
                                                 const float* __restrict__ bias_p) {
    __shared__ float As[2][TILE_FLOATS];
    __shared__ float Bs[2][TILE_FLOATS];

    const int b = blockIdx.z;
    const float* A  = Ag + (size_t)b * strideA;
    const float* Bm = Bg + (size_t)b * strideB;
    float*       C  = Cg + (size_t)b * strideC;

    const int rowBase = blockIdx.y * 128;
    const int colBase = blockIdx.x * 128;

    const int tid    = threadIdx.x;
    const int lane   = tid & 31;
    const int waveId = tid >> 5;
    const int wm     = waveId >> 1;   // 0..3 -> 32-row slab
    const int wn     = waveId & 1;    // 0..1 -> 64-col slab
    const int half   = lane >> 4;     // K-half selector per ISA A/B layout
    const int lrow   = lane & 15;

    v8f acc[2][4];
    #pragma unroll
    for (int i = 0; i < 2; ++i)
        #pragma unroll
        for (int j = 0; j < 4; ++j)
            acc[i][j] = (v8f)0.0f;

    // ---- async stage of one K-chunk into LDS buffer `buf` (8 ops/wave) ----
    auto issue_chunk = [&](int k0, int buf) {
        #pragma unroll
        for (int i = 0; i < 4; ++i) {            // A tile: 128 rows x 32 k
            int slot = tid + i * 256;            // 0..1023 float4 slots
            int r    = slot >> 3;
            int c4   = slot & 7;
            uint32_t goff = (uint32_t)((rowBase + r) * lda + k0 + c4 * 4) << 2;
            async_copy_b128(lds_addr_of(&As[buf][r * LDS_STRIDE + c4 * 4]), A, goff);
        }
        if (B_IS_KXN) {
            #pragma unroll
            for (int i = 0; i < 4; ++i) {        // U tile: 32 k-rows x 128 n
                int slot = tid + i * 256;
                int k    = slot >> 5;
                int n4   = slot & 31;
                uint32_t goff = (uint32_t)((k0 + k) * ldb + colBase + n4 * 4) << 2;
                async_copy_b128(lds_addr_of(&Bs[buf][k * KXN_STRIDE + n4 * 4]), Bm, goff);
            }
        } else {
            #pragma unroll
            for (int i = 0; i < 4; ++i) {        // dep tile: 128 rows x 32 k
                int slot = tid + i * 256;
                int r    = slot >> 3;
                int c4   = slot & 7;
                uint32_t goff = (uint32_t)((colBase + r) * ldb + k0 + c4 * 4) << 2;
                async_copy_b128(lds_addr_of(&Bs[buf][r * LDS_STRIDE + c4 * 4]), Bm, goff);
            }
        }
    };

    const int nChunks = K / BK;
    issue_chunk(0, 0);

    for (int c = 0; c < nChunks; ++c) {
        const int buf = c & 1;
        if (c + 1 < nChunks) {
            issue_chunk((c + 1) * BK, buf ^ 1);  // prefetch next chunk
            wait_async_le8();                    // chunk c's 8 ops complete (in-order)
        } else {
            wait_async_0();
        }
        __syncthreads();

        const float* __restrict__ Ab = As[buf];
        const float* __restrict__ Bb = Bs[buf];

        #pragma unroll
        for (int ks = 0; ks < BK / 4; ++ks) {
            const int kcol = ks * 4 + half * 2;  // lanes 0-15: K{0,1}; 16-31: K{2,3}

            v2f a0 = *(const v2f*)&Ab[(wm * 32 +  0 + lrow) * LDS_STRIDE + kcol];
            v2f a1 = *(const v2f*)&Ab[(wm * 32 + 16 + lrow) * LDS_STRIDE + kcol];

            v2f bf[4];
            #pragma unroll
            for (int nt = 0; nt < 4; ++nt) {
                const int n = wn * 64 + nt * 16 + lrow;
                if (B_IS_KXN) {
                    v2f t;
                    t[0] = Bb[(kcol + 0) * KXN_STRIDE + n];
                    t[1] = Bb[(kcol + 1) * KXN_STRIDE + n];
                    bf[nt] = t;
                } else {
                    bf[nt] = *(const v2f*)&Bb[n * LDS_STRIDE + kcol];
                }
            }

            #pragma unroll
            for (int nt = 0; nt < 4; ++nt) {
                acc[0][nt] = __builtin_amdgcn_wmma_f32_16x16x4_f32(
                    false, a0, false, bf[nt], (short)0, acc[0][nt], false, false);
                acc[1][nt] = __builtin_amdgcn_wmma_f32_16x16x4_f32(
                    false, a1, false, bf[nt], (short)0, acc[1][nt], false, false);
            }
        }
        __syncthreads();   // all waves done reading buf before it is refilled
    }

    // ---- epilogue + store (C layout: VGPR r -> row r + 8*half, col lrow) ----
    float bias = 0.0f;
    const float* hsb = nullptr;
    const float* dsb = nullptr;
    if (EPILOGUE) {
        bias = bias_p[0];
        hsb = hs + (size_t)b * S_DIM;
        dsb = ds + (size_t)b * S_DIM;
    }

    #pragma unroll
    for (int mt = 0; mt < 2; ++mt) {
        #pragma unroll
        for (int nt = 0; nt < 4; ++nt) {
            const int r0 = rowBase + wm * 32 + mt * 16 + half * 8;
            const int c  = colBase + wn * 64 + nt * 16 + lrow;
            float cadd = 0.0f;
            if (EPILOGUE) cadd = dsb[c] + bias;
            #pragma unroll
            for (int r = 0; r < 8; ++r) {
                float v = acc[mt][nt][r];
                if (EPILOGUE) v += hsb[r0 + r] + cadd;
                C[(size_t)(r0 + r) * ldc + c] = v;
            }
        }
    }
}

// ---------------------------------------------------------------------------
extern "C" void kernel_launch(void* const* d_in, const int* in_sizes, int n_in,
                              void* d_out, int out_size, void* d_ws, size_t ws_size,
                              hipStream_t stream) {
    const float* head   = (const float*)d_in[0];
    const float* dep    = (const float*)d_in[1];
    const float* edge_U = (const float*)d_in[2];
    const float* edge_W = (const float*)d_in[3];
    const float* edge_b = (const float*)d_in[4];
    float* out = (float*)d_out;

    // workspace layout
    float* Hh = (float*)d_ws;                                   // B*S*D
    float* hs = Hh + (size_t)B_DIM * S_DIM * D_DIM;             // B*S
    float* ds = hs + (size_t)B_DIM * S_DIM;                     // B*S

    // 1) rank-1 vectors
    {
        int waves = 2 * B_DIM * S_DIM;           // 32768 waves
        dim3 grid(waves / 8), block(256);
        rank1_kernel<<<grid, block, 0, stream>>>(head, dep, edge_W, hs, ds);
    }

    // 2) Hh = head (B*S x D) @ U (D x D)
    {
        dim3 grid(D_DIM / 128, (B_DIM * S_DIM) / 128, 1), block(256);
        wmma_gemm<true, false><<<grid, block, 0, stream>>>(
            head, edge_U, Hh,
            B_DIM * S_DIM, D_DIM, D_DIM,
            D_DIM, D_DIM, D_DIM,
            0L, 0L, 0L,
            nullptr, nullptr, nullptr);
    }

    // 3) out[b] = Hh[b] (S x D) @ dep[b]^T (D x S)  + hs + ds^T + b0
    {
        dim3 grid(S_DIM / 128, S_DIM / 128, B_DIM), block(256);
        wmma_gemm<false, true><<<grid, block, 0, stream>>>(
            Hh, dep, out,
            S_DIM, S_DIM, D_DIM,
            D_DIM, D_DIM, S_DIM,
            (long)S_DIM * D_DIM, (long)S_DIM * D_DIM, (long)S_DIM * S_DIM,
            hs, ds, edge_b);
    }
}